// Decoder_TRANSFORMER_14791867367496
// MI455X (gfx1250) — compile-verified
//
#include <hip/hip_runtime.h>
#include <hip/hip_bf16.h>

// Shapes from the reference
#define BS      32
#define NFRAMES 60
#define NJ      4096
#define NF      3
#define LATENT  256
#define CLIP    512

typedef __attribute__((ext_vector_type(2))) float v2f;
typedef __attribute__((ext_vector_type(4))) float v4f;
typedef __attribute__((ext_vector_type(8))) float v8f;

// Workspace layout (floats)
#define WS_ZP 0              // [32][256]   z_proj
#define WS_S  8192           // [60][32][3] s_t[b]
#define WS_G  13952          // [60][9]     G_t = A^t + N*B_t
#define WS_H  14492          // [60][3]     h_t = e*B_t
#define WS_C  14672          // [60][32][3] C_t[b]

// ---------------------------------------------------------------------------
// Kernel 1: z_proj = z @ W_clip + b_clip  (32x512x256 FP32 GEMM via WMMA)
// One wave per 16x16 output tile; K swept in steps of 4 with
// V_WMMA_F32_16X16X4_F32 (128 wmma ops per wave).
// A-frag layout (32-bit 16x4): lanes 0-15 hold M=lane, K=k,k+1 in v[0],v[1];
// lanes 16-31 hold M=lane-16, K=k+2,k+3. B-frag is the mirrored 4x16 layout.
// C/D: VGPR v holds row m0+v (lanes 0-15) / m0+v+8 (lanes 16-31), col n0+lane.
// ---------------------------------------------------------------------------
__global__ void zproj_wmma_kernel(const float* __restrict__ z,     // [32][512]
                                  const float* __restrict__ Wc,    // [512][256]
                                  const float* __restrict__ bc,    // [256]
                                  float* __restrict__ zp)          // [32][256]
{
    const int m0   = blockIdx.x * 16;     // 0 or 16
    const int n0   = blockIdx.y * 16;     // 0..240
    const int lane = threadIdx.x;         // wave32: 0..31
    const int half = lane >> 4;           // 0 | 1
    const int l    = lane & 15;

    v8f acc = {};
    #pragma unroll 4
    for (int k = 0; k < CLIP; k += 4) {
        const int ka = k + half * 2;
        v2f a, b;
        a.x = z[(m0 + l) * CLIP + ka + 0];
        a.y = z[(m0 + l) * CLIP + ka + 1];
        b.x = Wc[(ka + 0) * LATENT + n0 + l];
        b.y = Wc[(ka + 1) * LATENT + n0 + l];
        acc = __builtin_amdgcn_wmma_f32_16x16x4_f32(
                  /*neg_a=*/false, a, /*neg_b=*/false, b,
                  /*c_mod=*/(short)0, acc, /*reuse_a=*/false, /*reuse_b=*/false);
    }

    const int row = m0 + half * 8;
    const int col = n0 + l;
    const float bias = bc[col];
    #pragma unroll
    for (int v = 0; v < 8; ++v) {
        zp[(row + v) * LATENT + col] = acc[v] + bias;
    }
}

// ---------------------------------------------------------------------------
// Kernel 2 (tiny, one block): build the per-timestep 3x3 tables.
//   A = I + W_pe@W3, N = W_pe@W2, e = b_pe@(W2+W3)+b_final
//   S[t][b] = (emb[t]+zp[b]) @ W1
//   B_t = A*B_{t-1} + I ; P_t = P_{t-1}*A ; G_t = P_t + N*B_t ; H_t = e*B_t
//   C_t[b] = C_{t-1}[b]*A + S[t][b]
// ---------------------------------------------------------------------------
__global__ void prep_kernel(const float* __restrict__ W_pe,    // [3][256]
                            const float* __restrict__ b_pe,    // [256]
                            const float* __restrict__ emb,     // [60][256]
                            const float* __restrict__ W_final, // [768][3]
                            const float* __restrict__ b_final, // [3]
                            float* __restrict__ ws)
{
    float* zp = ws + WS_ZP;
    float* S  = ws + WS_S;
    float* G  = ws + WS_G;
    float* H  = ws + WS_H;
    float* C  = ws + WS_C;

    __shared__ float sA[9], sN[9], sE[3];
    const int tid = threadIdx.x;

    // Phase 1: 3x3 collapsed matrices (21 small dot products)
    if (tid < 9) {
        int i = tid / 3, j = tid % 3;
        float acc = 0.f;
        for (int l = 0; l < LATENT; ++l)
            acc += W_pe[i * LATENT + l] * W_final[(2 * LATENT + l) * NF + j];
        sA[tid] = acc + (i == j ? 1.f : 0.f);          // A = I + W_pe@W3
    } else if (tid < 18) {
        int q = tid - 9, i = q / 3, j = q % 3;
        float acc = 0.f;
        for (int l = 0; l < LATENT; ++l)
            acc += W_pe[i * LATENT + l] * W_final[(LATENT + l) * NF + j];
        sN[q] = acc;                                    // N = W_pe@W2
    } else if (tid < 21) {
        int j = tid - 18;
        float acc = b_final[j];
        for (int l = 0; l < LATENT; ++l)
            acc += b_pe[l] * (W_final[(LATENT + l) * NF + j] +
                              W_final[(2 * LATENT + l) * NF + j]);
        sE[j] = acc;                                    // e
    }

    // Phase 2: S[t][b][j] = (emb[t]+zp[b]) . W1[:,j]   (5760 length-256 dots)
    for (int idx = tid; idx < NFRAMES * BS * NF; idx += blockDim.x) {
        int t = idx / (BS * NF);
        int b = (idx / NF) % BS;
        int j = idx % NF;
        float acc = 0.f;
        for (int l = 0; l < LATENT; ++l)
            acc += (emb[t * LATENT + l] + zp[b * LATENT + l]) * W_final[l * NF + j];
        S[idx] = acc;
    }
    __syncthreads();

    // Phase 3a: serial 3x3 recurrences for G_t, H_t (thread 0)
    if (tid == 0) {
        float P[9], Bm[9];
        for (int q = 0; q < 9; ++q) { P[q] = (q % 4 == 0) ? 1.f : 0.f; Bm[q] = 0.f; }
        for (int q = 0; q < 9; ++q) G[q] = P[q];       // t=0: G=I
        for (int j = 0; j < 3; ++j) H[j] = 0.f;        // t=0: H=0
        for (int t = 1; t < NFRAMES; ++t) {
            float Pn[9], Bn[9];
            for (int i = 0; i < 3; ++i)
                for (int j = 0; j < 3; ++j) {
                    float pb = 0.f, pp = 0.f;
                    for (int k = 0; k < 3; ++k) {
                        pb += sA[i * 3 + k] * Bm[k * 3 + j];
                        pp += P[i * 3 + k] * sA[k * 3 + j];
                    }
                    Bn[i * 3 + j] = pb + (i == j ? 1.f : 0.f);
                    Pn[i * 3 + j] = pp;
                }
            for (int q = 0; q < 9; ++q) { P[q] = Pn[q]; Bm[q] = Bn[q]; }
            for (int i = 0; i < 3; ++i)
                for (int j = 0; j < 3; ++j) {
                    float nb = 0.f;
                    for (int k = 0; k < 3; ++k) nb += sN[i * 3 + k] * Bm[k * 3 + j];
                    G[t * 9 + i * 3 + j] = P[i * 3 + j] + nb;
                }
            for (int j = 0; j < 3; ++j) {
                float hb = 0.f;
                for (int i = 0; i < 3; ++i) hb += sE[i] * Bm[i * 3 + j];
                H[t * 3 + j] = hb;
            }
        }
    }

    // Phase 3b: C_t[b] recurrence, one thread per batch element
    if (tid < BS) {
        const int b = tid;
        float c0 = 0.f, c1 = 0.f, c2 = 0.f;
        C[(0 * BS + b) * 3 + 0] = 0.f;
        C[(0 * BS + b) * 3 + 1] = 0.f;
        C[(0 * BS + b) * 3 + 2] = 0.f;
        for (int t = 1; t < NFRAMES; ++t) {
            const float* s = &S[(t * BS + b) * 3];
            float n0 = c0 * sA[0] + c1 * sA[3] + c2 * sA[6] + s[0];
            float n1 = c0 * sA[1] + c1 * sA[4] + c2 * sA[7] + s[1];
            float n2 = c0 * sA[2] + c1 * sA[5] + c2 * sA[8] + s[2];
            c0 = n0; c1 = n1; c2 = n2;
            C[(t * BS + b) * 3 + 0] = c0;
            C[(t * BS + b) * 3 + 1] = c1;
            C[(t * BS + b) * 3 + 2] = c2;
        }
    }
}

// ---------------------------------------------------------------------------
// Kernel 3: out[b,j,f,t] = g[b,j]·G_t[:,f] + H_t[f] + C_t[b,f]
// Each thread emits 4 consecutive t (60 = 15 groups of 4, never crossing an
// f-run), as one 16B non-temporal store -> pure coalesced b128 write stream.
// 94 MB of stores ≈ 4 us at 23.3 TB/s is the roofline for this problem.
// ---------------------------------------------------------------------------
__global__ void decode_out_kernel(const float* __restrict__ grid,  // [32][4096][3]
                                  const float* __restrict__ ws,
                                  float* __restrict__ out)         // [32][4096][3][60]
{
    const float* G = ws + WS_G;
    const float* H = ws + WS_H;
    const float* C = ws + WS_C;

    const int total4 = BS * NJ * NF * (NFRAMES / 4);   // 5,898,240
    const int vid = blockIdx.x * blockDim.x + threadIdx.x;
    if (vid >= total4) return;

    const int q15 = vid % 15;            // which group-of-4 within the 60-run
    const int f   = (vid / 15) % NF;
    const int bj  = vid / (15 * NF);
    const int b   = bj >> 12;            // / NJ
    const int t0  = q15 * 4;

    const float g0 = grid[bj * 3 + 0];
    const float g1 = grid[bj * 3 + 1];
    const float g2 = grid[bj * 3 + 2];

    v4f r;
    #pragma unroll
    for (int q = 0; q < 4; ++q) {
        const int t = t0 + q;
        r[q] = g0 * G[t * 9 + 0 + f] +
               g1 * G[t * 9 + 3 + f] +
               g2 * G[t * 9 + 6 + f] +
               H[t * 3 + f] +
               C[(t * BS + b) * 3 + f];
    }

    // 16B-aligned: t0 % 4 == 0 and the f-run length (60) is a multiple of 4.
    v4f* dst = (v4f*)(out + ((long long)bj * NF + f) * NFRAMES + t0);
    __builtin_nontemporal_store(r, dst);
}

extern "C" void kernel_launch(void* const* d_in, const int* in_sizes, int n_in,
                              void* d_out, int out_size, void* d_ws, size_t ws_size,
                              hipStream_t stream) {
    (void)in_sizes; (void)n_in; (void)out_size; (void)ws_size;

    const float* z       = (const float*)d_in[0];  // [32][512]
    // d_in[1] = mask (bool, all ones) -- shape-only in the reference
    const float* grid    = (const float*)d_in[2];  // [32][4096][3]
    const float* W_pe    = (const float*)d_in[3];  // [3][256]
    const float* b_pe    = (const float*)d_in[4];  // [256]
    const float* W_clip  = (const float*)d_in[5];  // [512][256]
    const float* b_clip  = (const float*)d_in[6];  // [256]
    const float* emb     = (const float*)d_in[7];  // [60][256]
    const float* W_final = (const float*)d_in[8];  // [768][3]
    const float* b_final = (const float*)d_in[9];  // [3]

    float* ws  = (float*)d_ws;
    float* out = (float*)d_out;

    // 1) WMMA GEMM: z_proj = z @ W_clip + b_clip  (2 x 16 tiles, 1 wave each)
    zproj_wmma_kernel<<<dim3(2, 16), 32, 0, stream>>>(z, W_clip, b_clip, ws + WS_ZP);

    // 2) Collapse scan into per-t 3x3 tables (single small block)
    prep_kernel<<<1, 256, 0, stream>>>(W_pe, b_pe, emb, W_final, b_final, ws);

    // 3) Bandwidth-bound, float4 non-temporal coalesced output expansion
    const int total4  = BS * NJ * NF * (NFRAMES / 4);            // 5,898,240
    const int threads = 256;
    const int blocks  = (total4 + threads - 1) / threads;        // 23,040
    decode_out_kernel<<<blocks, threads, 0, stream>>>(grid, ws, out);
}